// TtFalconAttentionDecode_7559142441200
// MI455X (gfx1250) — compile-verified
//
#include <hip/hip_runtime.h>
#include <hip/hip_bf16.h>

// ---------------------------------------------------------------------------
// Falcon-7B attention decode step, batch=64, fp32, MI455X (gfx1250, wave32).
// Bandwidth-bound (~235 MB streamed vs ~7.7 GFLOP): use V_WMMA_F32_16X16X4_F32
// so issue slots go to memory instructions, stage activations in LDS, keep
// the whole 16x2048 score tile resident in the 320KB WGP LDS for an exact
// two-pass softmax.
// ---------------------------------------------------------------------------

#define HIDDEN   4544
#define NUM_HEADS 71
#define HEAD_DIM  64
#define MAX_POS  2048
#define BATCH      64
#define QKV_N    ((NUM_HEADS + 2) * HEAD_DIM)   // 4672

typedef float v2f __attribute__((ext_vector_type(2)));
typedef float v8f __attribute__((ext_vector_type(8)));

__device__ __forceinline__ v8f wmma4(v2f a, v2f b, v8f c) {
  // D[16x16] += A[16x4] * B[4x16], fp32. 8-arg form:
  // (neg_a, A, neg_b, B, c_mod, C, reuse_a, reuse_b)
  return __builtin_amdgcn_wmma_f32_16x16x4_f32(false, a, false, b, (short)0, c,
                                               false, false);
}

// ---------------------------------------------------------------------------
// C[64 x N] = A[64 x K] * B[K x N], fp32 WMMA.
// 8 waves / WG; wave w owns columns n0..n0+15 and all 4 M-tiles (M=64).
// A chunk (64 x 64) staged in LDS with padded stride; B streamed once with
// an unconditional prefetch 64 rows ahead (speculative, silently dropped if
// OOB).  All B addressing is strength-reduced 32-bit element offsets so the
// hot loop is add+load, not v_mul_u64 chains.
// ---------------------------------------------------------------------------
#define LDS_A_STRIDE 68

__global__ __launch_bounds__(256) void gemm64_wmma(
    const float* __restrict__ A, const float* __restrict__ B,
    float* __restrict__ C, int K, int N) {
  __shared__ float lA[64 * LDS_A_STRIDE];
  const int tid  = threadIdx.x;
  const int wave = tid >> 5;
  const int lane = tid & 31;
  const int nl   = lane & 15;
  const int hh   = lane >> 4;

  int n0 = blockIdx.x * 128 + wave * 16;
  if (n0 > N - 16) n0 = N - 16;  // N % 16 == 0; duplicated work is benign

  const unsigned uN = (unsigned)N;
  v8f acc0 = {}, acc1 = {}, acc2 = {}, acc3 = {};

  // loop-carried 32-bit element offset into B (fits: B <= 85 MB)
  unsigned boff = (unsigned)(2 * hh) * uN + (unsigned)(n0 + nl);

  for (int kc = 0; kc < K; kc += 64) {
    __syncthreads();
    // Cooperative load of A[0:64][kc:kc+64] into LDS (1024 float4 / 256 thr).
#pragma unroll
    for (int i = 0; i < 4; ++i) {
      unsigned idx = (unsigned)tid * 4u + (unsigned)i;
      unsigned row = idx >> 4;
      unsigned f4  = idx & 15u;
      float4 v = *(const float4*)(A + row * (unsigned)K + (unsigned)kc + f4 * 4u);
      *(float4*)&lA[row * LDS_A_STRIDE + f4 * 4u] = v;
    }
    __syncthreads();

#pragma unroll 4
    for (int s = 0; s < 16; ++s) {
      const int kl = 4 * s + 2 * hh;
      v2f bv;
      bv.x = B[boff];
      bv.y = B[boff + uN];
      // stream-ahead by 64 rows; speculative prefetch, dropped if OOB
      __builtin_prefetch(&B[boff + 64u * uN], 0, 3);
      v2f a0 = *(const v2f*)&lA[(nl +  0) * LDS_A_STRIDE + kl];
      v2f a1 = *(const v2f*)&lA[(nl + 16) * LDS_A_STRIDE + kl];
      v2f a2 = *(const v2f*)&lA[(nl + 32) * LDS_A_STRIDE + kl];
      v2f a3 = *(const v2f*)&lA[(nl + 48) * LDS_A_STRIDE + kl];
      acc0 = wmma4(a0, bv, acc0);
      acc1 = wmma4(a1, bv, acc1);
      acc2 = wmma4(a2, bv, acc2);
      acc3 = wmma4(a3, bv, acc3);
      boff += 4u * uN;
    }
  }

  const unsigned cb = (unsigned)(n0 + nl);
#pragma unroll
  for (int r = 0; r < 8; ++r) {
    unsigned m = (unsigned)(r + 8 * hh);
    C[(m)       * uN + cb] = acc0[r];
    C[(m + 16u) * uN + cb] = acc1[r];
    C[(m + 32u) * uN + cb] = acc2[r];
    C[(m + 48u) * uN + cb] = acc3[r];
  }
}

// ---------------------------------------------------------------------------
// RoPE on q (71 heads) and k; split v. inv_freq[j] = 10000^{-j/32}.
// ---------------------------------------------------------------------------
__global__ __launch_bounds__(256) void rope_split(
    const float* __restrict__ fused, float* __restrict__ q_rot,
    float* __restrict__ k_new, float* __restrict__ v_new,
    const int* __restrict__ p_ptr) {
  const int b = blockIdx.x;
  const float p = (float)(*p_ptr);
  const float* fb = fused + (size_t)b * QKV_N;
  for (int idx = threadIdx.x; idx < QKV_N; idx += blockDim.x) {
    int h = idx >> 6, d = idx & 63;
    float x = fb[idx];
    if (h <= NUM_HEADS) {  // q heads + k
      int j = d & 31;
      float ang = p * __expf(-(float)j * 0.28782313662425572f);  // ln(1e4)/32
      float c, s;
      __sincosf(ang, &s, &c);
      float other = (d < 32) ? -fb[h * 64 + d + 32] : fb[h * 64 + d - 32];
      float r = x * c + other * s;
      if (h < NUM_HEADS) q_rot[((size_t)b * NUM_HEADS + h) * 64 + d] = r;
      else               k_new[(size_t)b * 64 + d] = r;
    } else {
      v_new[(size_t)b * 64 + d] = x;
    }
  }
}

// ---------------------------------------------------------------------------
// Build the patched 16-row tail tile (cache rows below p, new k/v at p, zeros
// above) so the attention kernel never needs to mutate the input caches.
// ---------------------------------------------------------------------------
__global__ __launch_bounds__(256) void patch_tail(
    const float* __restrict__ cache_k, const float* __restrict__ cache_v,
    const float* __restrict__ k_new, const float* __restrict__ v_new,
    float* __restrict__ ktail, float* __restrict__ vtail,
    const int* __restrict__ p_ptr) {
  const int b = blockIdx.x;
  const int p = *p_ptr;
  const int t0 = p & ~15;
  for (int idx = threadIdx.x; idx < 16 * 64; idx += blockDim.x) {
    int tr = idx >> 6, d = idx & 63;
    int t = t0 + tr;
    float kv = 0.f, vv = 0.f;
    if (t < p) {
      kv = cache_k[((size_t)b * MAX_POS + t) * 64 + d];
      vv = cache_v[((size_t)b * MAX_POS + t) * 64 + d];
    } else if (t == p) {
      kv = k_new[(size_t)b * 64 + d];
      vv = v_new[(size_t)b * 64 + d];
    }
    ktail[((size_t)b * 16 + tr) * 64 + d] = kv;
    vtail[((size_t)b * 16 + tr) * 64 + d] = vv;
  }
}

// ---------------------------------------------------------------------------
// Attention: one WG per (batch, 16-head tile); 8 waves.
// Pass 1: S^T tiles via WMMA (A = K rows, B = q^T) -> softmax axis is along
//         VGPRs + lane-halves; per-head max via shfl_xor(16) + LDS reduce.
//         All 16x2048 scores kept in LDS (136 KB of the 320 KB WGP LDS).
// Pass 2: P = exp(s-m) re-read in WMMA A-layout from LDS; ctx via WMMA;
//         cross-wave combine with ds_add_f32.
// ---------------------------------------------------------------------------
#define S_STRIDE 17
#define ATTN_LDS_FLOATS (MAX_POS * S_STRIDE + 128 + 16 + 16 + 16 * 65)
#define ATTN_LDS_BYTES  (ATTN_LDS_FLOATS * 4)

__global__ __launch_bounds__(256) void attn_kernel(
    const float* __restrict__ q_rot, const float* __restrict__ cache_k,
    const float* __restrict__ cache_v, const float* __restrict__ ktail,
    const float* __restrict__ vtail, const float* __restrict__ amask,
    float* __restrict__ ctx, const int* __restrict__ p_ptr) {
  extern __shared__ float lds[];
  float* S      = lds;                       // [2048][17]
  float* red    = S + MAX_POS * S_STRIDE;    // [8][16]
  float* mfin   = red + 128;                 // [16]
  float* lfin   = mfin + 16;                 // [16]
  float* ctxacc = lfin + 16;                 // [16][65]

  const int b  = blockIdx.x / 5;
  const int ht = blockIdx.x % 5;
  const int h0 = ht * 16;
  const int p  = *p_ptr;
  const int ptile = p & ~15;

  const int tid  = threadIdx.x;
  const int wave = tid >> 5;
  const int lane = tid & 31;
  const int nl   = lane & 15;
  const int hh   = lane >> 4;
  const float scale = 0.125f;  // 1/sqrt(64)

  for (int i = tid; i < 16 * 65; i += 256) ctxacc[i] = 0.f;

  // Q^T operand registers: lane column = local head nl; reg pair covers
  // k = 4s + 2*hh + {0,1}.  Padded heads (>=71) read as zero.
  v2f q[16];
  {
    const int head = h0 + nl;
    const bool hv  = head < NUM_HEADS;
    const float* qp =
        q_rot + (size_t)(hv ? (b * NUM_HEADS + head) : 0) * 64 + 2 * hh;
#pragma unroll
    for (int s = 0; s < 16; ++s) {
      v2f t = *(const v2f*)&qp[4 * s];  // 8B-aligned b64 load
      q[s].x = hv ? t.x : 0.f;
      q[s].y = hv ? t.y : 0.f;
    }
  }

  // ---------------- Pass 1: scores ----------------
  float mrun = -3.0e38f;
  const float* Kb = cache_k + (size_t)b * MAX_POS * 64;
  for (int c = 0; c < 16; ++c) {
    const int t0 = c * 128 + wave * 16;
    const float* kp = ((t0 == ptile)
                           ? (ktail + ((size_t)b * 16 + nl) * 64)
                           : (Kb + (size_t)(t0 + nl) * 64)) + 2 * hh;
    v8f sc = {};
#pragma unroll
    for (int s = 0; s < 16; ++s) {
      v2f a = *(const v2f*)&kp[4 * s];  // K row pair, b64 load
      sc = wmma4(a, q[s], sc);  // S^T tile: row t = r + 8*hh, col head = nl
    }
    float lm = -3.0e38f;
#pragma unroll
    for (int r = 0; r < 8; ++r) {
      int t = t0 + r + 8 * hh;
      float v = sc[r] * scale + amask[(size_t)b * MAX_POS + t];
      if (t > p) v = -1.0e9f;
      lm = fmaxf(lm, v);
      S[t * S_STRIDE + nl] = v;
    }
    mrun = fmaxf(mrun, lm);
  }
  mrun = fmaxf(mrun, __shfl_xor(mrun, 16, 32));
  if (lane < 16) red[wave * 16 + nl] = mrun;
  __syncthreads();
  if (tid < 16) {
    float m = red[tid];
#pragma unroll
    for (int w = 1; w < 8; ++w) m = fmaxf(m, red[w * 16 + tid]);
    mfin[tid] = m;
  }
  __syncthreads();

  // ---------------- Pass 2: exp, sum, ctx ----------------
  const float mh = mfin[nl];
  float lsum = 0.f;
  v8f c0 = {}, c1 = {}, c2 = {}, c3 = {};
  const float* Vb = cache_v + (size_t)b * MAX_POS * 64;
  for (int c = 0; c < 16; ++c) {
    const int t0 = c * 128 + wave * 16;
    const bool tail = (t0 == ptile);
    const float* vrow = tail ? (vtail + (size_t)b * 16 * 64)
                             : (Vb + (size_t)t0 * 64);
    const float* slab = S + (t0 + 2 * hh) * S_STRIDE + nl;
#pragma unroll
    for (int s = 0; s < 16; ++s) {
      v2f a;  // P in A-layout: row head = nl, k = t
      a.x = __expf(slab[(4 * s) * S_STRIDE] - mh);
      a.y = __expf(slab[(4 * s + 1) * S_STRIDE] - mh);
      lsum += a.x + a.y;
      const float* vp0 = vrow + (4 * s + 2 * hh) * 64;
      const float* vp1 = vp0 + 64;
      v2f bv;
      bv.x = vp0[nl];       bv.y = vp1[nl];       c0 = wmma4(a, bv, c0);
      bv.x = vp0[16 + nl];  bv.y = vp1[16 + nl];  c1 = wmma4(a, bv, c1);
      bv.x = vp0[32 + nl];  bv.y = vp1[32 + nl];  c2 = wmma4(a, bv, c2);
      bv.x = vp0[48 + nl];  bv.y = vp1[48 + nl];  c3 = wmma4(a, bv, c3);
    }
  }
  lsum += __shfl_xor(lsum, 16, 32);
  __syncthreads();
  if (lane < 16) red[wave * 16 + nl] = lsum;
  __syncthreads();
  if (tid < 16) {
    float l = 0.f;
#pragma unroll
    for (int w = 0; w < 8; ++w) l += red[w * 16 + tid];
    lfin[tid] = l;
  }

  // ctx C-layout: row head = r + 8*hh, col d = nl (+16/32/48 per tile)
#pragma unroll
  for (int r = 0; r < 8; ++r) {
    int h = r + 8 * hh;
    atomicAdd(&ctxacc[h * 65 + nl],      c0[r]);
    atomicAdd(&ctxacc[h * 65 + 16 + nl], c1[r]);
    atomicAdd(&ctxacc[h * 65 + 32 + nl], c2[r]);
    atomicAdd(&ctxacc[h * 65 + 48 + nl], c3[r]);
  }
  __syncthreads();

  for (int i = tid; i < 16 * 64; i += 256) {
    int h = i >> 6, d = i & 63;
    int head = h0 + h;
    if (head < NUM_HEADS)
      ctx[(size_t)b * HIDDEN + head * 64 + d] = ctxacc[h * 65 + d] / lfin[h];
  }
}

// ---------------------------------------------------------------------------
extern "C" void kernel_launch(void* const* d_in, const int* in_sizes, int n_in,
                              void* d_out, int out_size, void* d_ws,
                              size_t ws_size, hipStream_t stream) {
  const float* hs    = (const float*)d_in[0];  // [1,1,64,4544]
  const float* Wqkv  = (const float*)d_in[1];  // [4544,4672]
  const float* Wd    = (const float*)d_in[2];  // [4544,4544]
  const float* ck    = (const float*)d_in[3];  // [64,1,2048,64]
  const float* cv    = (const float*)d_in[4];  // [64,1,2048,64]
  const float* amask = (const float*)d_in[5];  // [64,1,1,2048]
  const int*   pp    = (const int*)d_in[6];    // scalar
  float* out = (float*)d_out;

  float* ws    = (float*)d_ws;
  float* fused = ws;                                   // 64*4672
  float* qrot  = fused + (size_t)BATCH * QKV_N;        // 64*71*64
  float* knew  = qrot + (size_t)BATCH * NUM_HEADS * 64;
  float* vnew  = knew + (size_t)BATCH * 64;
  float* ktail = vnew + (size_t)BATCH * 64;            // 64*16*64
  float* vtail = ktail + (size_t)BATCH * 16 * 64;
  float* ctx   = vtail + (size_t)BATCH * 16 * 64;      // 64*4544

  hipFuncSetAttribute(reinterpret_cast<const void*>(attn_kernel),
                      hipFuncAttributeMaxDynamicSharedMemorySize,
                      ATTN_LDS_BYTES);

  // 1) fused = hs @ W_qkv
  gemm64_wmma<<<dim3((QKV_N + 127) / 128), 256, 0, stream>>>(
      hs, Wqkv, fused, HIDDEN, QKV_N);
  // 2) RoPE + split
  rope_split<<<dim3(BATCH), 256, 0, stream>>>(fused, qrot, knew, vnew, pp);
  // 3) patched tail tiles (caches stay read-only)
  patch_tail<<<dim3(BATCH), 256, 0, stream>>>(ck, cv, knew, vnew, ktail, vtail,
                                              pp);
  // 4) attention
  attn_kernel<<<dim3(BATCH * 5), 256, ATTN_LDS_BYTES, stream>>>(
      qrot, ck, cv, ktail, vtail, amask, ctx, pp);
  // 5) out = ctx @ W_dense
  gemm64_wmma<<<dim3((HIDDEN + 127) / 128), 256, 0, stream>>>(
      ctx, Wd, out, HIDDEN, HIDDEN);
}